// SelectivePerLevel_48730698940786
// MI455X (gfx1250) — compile-verified
//
#include <hip/hip_runtime.h>
#include <hip/hip_bf16.h>

typedef float v2f __attribute__((ext_vector_type(2)));
typedef float v8f __attribute__((ext_vector_type(8)));

#define HW    128
#define CCH   256
#define NB    2
#define NWIN  3
#define NEG_SLOPE 0.2f
#define EPSV  1e-12f

// ---------------- async global->LDS helpers (CDNA5) ----------------
#if defined(__has_builtin)
#if __has_builtin(__builtin_amdgcn_global_load_async_to_lds_b128)
#define HAVE_ASYNC_LDS 1
#endif
#if __has_builtin(__builtin_amdgcn_s_wait_asynccnt)
#define HAVE_WAIT_ASYNC 1
#endif
#endif

// b128 payload type as required by the builtin: int vector_size(16)
typedef int b128_t __attribute__((vector_size(16)));
typedef __attribute__((address_space(1))) b128_t* gas_b128_p;
typedef __attribute__((address_space(3))) b128_t* las_b128_p;

__device__ __forceinline__ void async_copy_b128(const float* gsrc, float* lds_dst) {
#ifdef HAVE_ASYNC_LDS
    gas_b128_p g = (gas_b128_p)(unsigned long long)(gsrc);
    las_b128_p l = (las_b128_p)(unsigned)(unsigned long long)(lds_dst);
    __builtin_amdgcn_global_load_async_to_lds_b128(g, l, 0, 0);
#else
    *(float4*)lds_dst = *(const float4*)gsrc;
#endif
}

__device__ __forceinline__ void wait_async_lds() {
#ifdef HAVE_ASYNC_LDS
#ifdef HAVE_WAIT_ASYNC
    __builtin_amdgcn_s_wait_asynccnt(0);
#else
    asm volatile("s_wait_asynccnt 0" ::: "memory");
#endif
#endif
}

// ---------------- kernel 1: windowed mean/std -> pooled [3][2][512] ----------------
__global__ __launch_bounds__(128)
void spl_window_stats_kernel(const float* __restrict__ feature, float* __restrict__ pooled) {
    const int planes_per_win = NB * CCH;
    const int win = blockIdx.x / planes_per_win;
    const int rem = blockIdx.x % planes_per_win;
    const int n   = rem / CCH;
    const int c   = rem % CCH;

    const int Ks[NWIN] = {3, 34, 65};
    const int K = Ks[win];
    int s = (K - 1) / 2; if (s < 1) s = 1;
    const int h = (HW - K) / s + 1;   // windows per dim (h == w)

    __shared__ float tile[HW * HW];   // 64 KB plane
    __shared__ float redM, redS;

    const float* src = feature + (size_t)(n * CCH + c) * (HW * HW);
    const int t = threadIdx.x;

    // stage plane into LDS (async b128 path on CDNA5)
    for (int j = t; j < HW * HW / 4; j += blockDim.x) {
        async_copy_b128(src + j * 4, tile + j * 4);
    }
    wait_async_lds();
    if (t == 0) { redM = 0.0f; redS = 0.0f; }
    __syncthreads();

    float tm = 0.0f, ts = 0.0f;
    if (t < h) {                      // thread owns one output column
        const int x0 = t * s;
        const float invn = 1.0f / (float)(K * K);
        for (int v = 0; v < h; ++v) {
            float ssum = 0.0f, ssq = 0.0f;
            const int y0 = v * s;
            for (int y = y0; y < y0 + K; ++y) {
                const float* row = tile + y * HW + x0;
                for (int x = 0; x < K; ++x) {
                    const float val = row[x];
                    ssum += val;
                    ssq  += val * val;
                }
            }
            const float mean = ssum * invn;
            float var = ssq * invn - mean * mean;
            var = fmaxf(var, 0.0f);
            tm += mean;
            ts += sqrtf(var + EPSV);
        }
    }
    atomicAdd(&redM, tm);             // ds_add_f32
    atomicAdd(&redS, ts);
    __syncthreads();

    if (t == 0) {
        const float inv = 1.0f / (float)(h * h);
        float* p = pooled + (size_t)(win * NB + n) * (2 * CCH);
        p[c]       = redM * inv;      // mean channels [0,256)
        p[CCH + c] = redS * inv;      // std  channels [256,512)
    }
}

// ---------------- WMMA f32 16x16x4 dense layer: out = leaky(in @ W^T + b) ----------------
// xin/xout: [16][512] LDS buffers (row stride 512); only rows 0..1 carry valid data,
// padded rows never contaminate valid rows (row-wise independence of GEMM).
__device__ __forceinline__ void mlp_layer_wmma(const float* __restrict__ xin,
                                               float* __restrict__ xout,
                                               const float* __restrict__ W,
                                               const float* __restrict__ B,
                                               int din, int dout, bool relu) {
    const int lane = threadIdx.x;     // 32 threads, one wave
    const int half = lane >> 4;       // 0: K=0,1   1: K=2,3
    const int m16  = lane & 15;       // A row (lanes 0-15) / B col (both halves)

    for (int tile = 0; tile < dout / 16; ++tile) {
        v8f acc = {0.f, 0.f, 0.f, 0.f, 0.f, 0.f, 0.f, 0.f};
        const int nrow = tile * 16 + m16;           // output feature index for this lane
        const float* wr = W + (size_t)nrow * din;   // W is [dout][din] row-major
        for (int k0 = 0; k0 < din; k0 += 4) {
            const int ka = k0 + 2 * half;
            v2f a, b;
            a.x = xin[m16 * 512 + ka];
            a.y = xin[m16 * 512 + ka + 1];
            b.x = wr[ka];                           // B[k][n] = W[n][k]
            b.y = wr[ka + 1];
            acc = __builtin_amdgcn_wmma_f32_16x16x4_f32(
                      false, a, false, b, (short)0, acc, false, false);
        }
        const float bias = B[nrow];
#pragma unroll
        for (int r = 0; r < 8; ++r) {
            const int mrow = (lane < 16) ? r : (8 + r);   // C/D layout: VGPR r -> M=r / M=8+r
            float v = acc[r] + bias;
            if (relu) v = (v > 0.0f) ? v : NEG_SLOPE * v;
            xout[mrow * 512 + nrow] = v;
        }
    }
}

// ---------------- kernel 2: per-window MLP + BCE loss ----------------
__global__ __launch_bounds__(32)
void spl_mlp_loss_kernel(const float* __restrict__ pooled,
                         const float* __restrict__ W1, const float* __restrict__ b1,
                         const float* __restrict__ W2, const float* __restrict__ b2,
                         const float* __restrict__ W3, const float* __restrict__ b3,
                         const float* __restrict__ W4, const float* __restrict__ b4,
                         const int* __restrict__ label,
                         float* __restrict__ out) {
    const int win  = blockIdx.x;
    const int lane = threadIdx.x;

    __shared__ float xa[16 * 512];
    __shared__ float xb[16 * 512];

    for (int i = lane; i < 16 * 512; i += 32) { xa[i] = 0.0f; xb[i] = 0.0f; }
    __syncthreads();

    const float* p = pooled + (size_t)win * NB * (2 * CCH);
    for (int i = lane; i < NB * 512; i += 32) {
        const int m = i / 512, k = i % 512;
        xa[m * 512 + k] = p[i];
    }
    __syncthreads();

    mlp_layer_wmma(xa, xb, W1 + (size_t)win * 256 * 512, b1 + win * 256, 512, 256, true);
    __syncthreads();
    mlp_layer_wmma(xb, xa, W2 + (size_t)win * 256 * 256, b2 + win * 256, 256, 256, true);
    __syncthreads();
    mlp_layer_wmma(xa, xb, W3 + (size_t)win * 128 * 256, b3 + win * 128, 256, 128, true);
    __syncthreads();

    if (lane == 0) {
        const float* w4 = W4 + (size_t)win * 128;
        const float bias4 = b4[win];
        const float yv = (float)label[0];
        float acc = 0.0f;
        for (int m = 0; m < NB; ++m) {
            float l = bias4;
            for (int k = 0; k < 128; ++k) l += xb[m * 512 + k] * w4[k];
            // stable softplus
            const float sp = fmaxf(l, 0.0f) + log1pf(expf(-fabsf(l)));
            acc += sp - l * yv;
        }
        atomicAdd(out, acc * (1.0f / (float)NB));
    }
}

__global__ void spl_zero_out_kernel(float* out) {
    if (threadIdx.x == 0 && blockIdx.x == 0) out[0] = 0.0f;
}

extern "C" void kernel_launch(void* const* d_in, const int* in_sizes, int n_in,
                              void* d_out, int out_size, void* d_ws, size_t ws_size,
                              hipStream_t stream) {
    const float* feature = (const float*)d_in[0];
    const float* W1 = (const float*)d_in[1];
    const float* b1 = (const float*)d_in[2];
    const float* W2 = (const float*)d_in[3];
    const float* b2 = (const float*)d_in[4];
    const float* W3 = (const float*)d_in[5];
    const float* b3 = (const float*)d_in[6];
    const float* W4 = (const float*)d_in[7];
    const float* b4 = (const float*)d_in[8];
    const int* label = (const int*)d_in[9];
    float* out = (float*)d_out;

    float* pooled = (float*)d_ws;     // [3][2][512] floats = 12 KB

    spl_zero_out_kernel<<<1, 1, 0, stream>>>(out);
    spl_window_stats_kernel<<<NWIN * NB * CCH, 128, 0, stream>>>(feature, pooled);
    spl_mlp_loss_kernel<<<NWIN, 32, 0, stream>>>(pooled, W1, b1, W2, b2, W3, b3, W4, b4,
                                                 label, out);
}